// GGNNEncoder_20315195310533
// MI455X (gfx1250) — compile-verified
//
#include <hip/hip_runtime.h>

// GGNN encoder, MI455X (gfx1250, wave32).
// Key idea: msg[e] = A[e] @ h[src[e]] with A[e] = reshape(W_e @ ef[e] + b_e)
// factorizes into msg = kron(h[src[e]], [ef[e],1]) @ W2^T, a shared GEMM
// [E,544]x[544,32]. This avoids materializing the 655MB A tensor (HBM-bound)
// and turns the dominant work into fp32 WMMA (v_wmma_f32_16x16x4_f32).

#define HH    32      // hidden
#define EDIM  16      // edge feat dim
#define KF    544     // 32 * 17 (ef augmented with constant 1 to fold b_e)
#define NCH   136     // KF / 4 K-chunks per WMMA accumulation
#define BFRAG_ELEMS (2 * NCH * 32)        // float2 elements
#define BFRAG_BYTES (BFRAG_ELEMS * 8)     // 69632 B

typedef float v2f __attribute__((ext_vector_type(2)));
typedef float v8f __attribute__((ext_vector_type(8)));

// ---------------------------------------------------------------------------
// Build B-fragments of W2^T in WMMA 4x16 B layout, once.
// W2[i][k*17+d] = (d<16) ? W_e[(i*32+k)*16+d] : b_e[i*32+k]
// B-fragment (per chunk c, per lane): lanes 0-15 hold cols n=(lane&15)+16t with
// K rows {4c,4c+1}; lanes 16-31 hold K rows {4c+2,4c+3}.
// ---------------------------------------------------------------------------
__device__ __forceinline__ float w2val(const float* __restrict__ W_e,
                                       const float* __restrict__ b_e,
                                       int kf, int i) {
    int k = kf / 17;
    int d = kf % 17;
    return (d < 16) ? W_e[(i * HH + k) * EDIM + d] : b_e[i * HH + k];
}

__global__ void ggnn_build_bfrag(const float* __restrict__ W_e,
                                 const float* __restrict__ b_e,
                                 float2* __restrict__ bfrag) {
    int idx = blockIdx.x * blockDim.x + threadIdx.x;
    if (idx >= BFRAG_ELEMS) return;
    int lane = idx & 31;
    int c    = (idx >> 5) % NCH;
    int t    = idx / (NCH * 32);
    int n_i  = (lane & 15) + 16 * t;          // output channel
    int kk   = (lane < 16) ? 0 : 2;
    int kf0  = 4 * c + kk;
    float2 v;
    v.x = w2val(W_e, b_e, kf0,     n_i);
    v.y = w2val(W_e, b_e, kf0 + 1, n_i);
    bfrag[idx] = v;
}

// ---------------------------------------------------------------------------
// Zero the message accumulator (ws is poisoned by the harness).
// ---------------------------------------------------------------------------
__global__ void ggnn_zero_m(float* __restrict__ m, int n) {
    int i = blockIdx.x * blockDim.x + threadIdx.x;
    if (i < n) m[i] = 0.0f;
}

// ---------------------------------------------------------------------------
// h0 = node_feat @ W_np.T + b_np   ([N,64]x[64,32], ~82 MFLOP: VALU kernel)
// ---------------------------------------------------------------------------
__global__ __launch_bounds__(256) void ggnn_node_proj(
    const float* __restrict__ nf, const float* __restrict__ W_np,
    const float* __restrict__ b_np, float* __restrict__ h, int Nn) {
    __shared__ float sW[HH * 64];
    __shared__ float sb[HH];
    for (int i = threadIdx.x; i < HH * 64; i += blockDim.x) sW[i] = W_np[i];
    for (int i = threadIdx.x; i < HH;     i += blockDim.x) sb[i] = b_np[i];
    __syncthreads();
    int v = blockIdx.x * blockDim.x + threadIdx.x;
    if (v >= Nn) return;
    float x[64];
    const float4* xp = (const float4*)(nf + (size_t)v * 64);
    #pragma unroll
    for (int q = 0; q < 16; ++q) {
        float4 t = xp[q];
        x[4*q+0] = t.x; x[4*q+1] = t.y; x[4*q+2] = t.z; x[4*q+3] = t.w;
    }
    #pragma unroll
    for (int i = 0; i < HH; ++i) {
        float acc = sb[i];
        #pragma unroll
        for (int k = 0; k < 64; ++k) acc = fmaf(sW[i * 64 + k], x[k], acc);
        h[(size_t)v * HH + i] = acc;
    }
}

// ---------------------------------------------------------------------------
// Message pass: one wave per 16-edge tile.
// C[16 edges x 32 ch] = O[16 x 544] x W2t[544 x 32] via 2x136 f32 WMMAs,
// O built on the fly in registers, B-fragments streamed from LDS,
// scatter-add into m via hardware f32 atomics.
// ---------------------------------------------------------------------------
__global__ __launch_bounds__(256) void ggnn_message(
    const float* __restrict__ h,          // [N,32]
    const float* __restrict__ edge_feat,  // [E,16]
    const int*   __restrict__ src,        // [E]
    const int*   __restrict__ dst,        // [E]
    const float2* __restrict__ bfrag_g,   // [2][136][32]
    float* __restrict__ m,                // [N,32] accumulator
    int Ee) {
    __shared__ v2f sB[BFRAG_ELEMS];
    for (int i = threadIdx.x; i < BFRAG_ELEMS; i += blockDim.x) {
        float2 t = bfrag_g[i];
        v2f u; u.x = t.x; u.y = t.y;
        sB[i] = u;
    }
    __syncthreads();

    const int wave = threadIdx.x >> 5;
    const int lane = threadIdx.x & 31;
    const int tile = blockIdx.x * (blockDim.x >> 5) + wave;
    const int e0   = tile * 16;
    if (e0 >= Ee) return;                     // wave-uniform: EXEC stays full

    const int  r   = lane & 15;               // edge row this lane feeds
    const bool lo  = lane < 16;               // K 0,1 half vs K 2,3 half
    const int  e   = e0 + r;
    const int  s   = src[e];

    // per-lane row data: ef17 (16 + constant 1) and hs (32)
    float ef[17];
    {
        const float4* p = (const float4*)(edge_feat + (size_t)e * EDIM);
        #pragma unroll
        for (int q = 0; q < 4; ++q) {
            float4 t = p[q];
            ef[4*q+0] = t.x; ef[4*q+1] = t.y; ef[4*q+2] = t.z; ef[4*q+3] = t.w;
        }
        ef[16] = 1.0f;
    }
    float hs[HH];
    {
        const float4* p = (const float4*)(h + (size_t)s * HH);
        #pragma unroll
        for (int q = 0; q < 8; ++q) {
            float4 t = p[q];
            hs[4*q+0] = t.x; hs[4*q+1] = t.y; hs[4*q+2] = t.z; hs[4*q+3] = t.w;
        }
    }

    v8f c0 = {0.f, 0.f, 0.f, 0.f, 0.f, 0.f, 0.f, 0.f};
    v8f c1 = {0.f, 0.f, 0.f, 0.f, 0.f, 0.f, 0.f, 0.f};

    #pragma unroll
    for (int c = 0; c < NCH; ++c) {
        const int f = 4 * c;                  // compile-time after unroll
        // A fragment: lanes 0-15 supply O[r][f..f+1], lanes 16-31 O[r][f+2..f+3]
        float axl = hs[(f+0)/17] * ef[(f+0)%17];
        float ayl = hs[(f+1)/17] * ef[(f+1)%17];
        float axh = hs[(f+2)/17] * ef[(f+2)%17];
        float ayh = hs[(f+3)/17] * ef[(f+3)%17];
        v2f a;
        a.x = lo ? axl : axh;
        a.y = lo ? ayl : ayh;
        v2f b0 = sB[(0 * NCH + c) * 32 + lane];
        v2f b1 = sB[(1 * NCH + c) * 32 + lane];
        c0 = __builtin_amdgcn_wmma_f32_16x16x4_f32(false, a, false, b0,
                                                   (short)0, c0, false, false);
        c1 = __builtin_amdgcn_wmma_f32_16x16x4_f32(false, a, false, b1,
                                                   (short)0, c1, false, false);
    }

    // Scatter: C VGPR j holds M=j (lanes 0-15) / M=j+8 (lanes 16-31), N=lane&15.
    const int n0    = lane & 15;
    const int rbase = lo ? 0 : 8;
    #pragma unroll
    for (int j = 0; j < 8; ++j) {
        int dj = dst[e0 + rbase + j];
        unsafeAtomicAdd(&m[(size_t)dj * HH + n0],      c0[j]);
        unsafeAtomicAdd(&m[(size_t)dj * HH + n0 + 16], c1[j]);
    }
}

// ---------------------------------------------------------------------------
// GRU cell (torch gate order r,z,n), one thread per node; weights in LDS.
// Also re-zeroes m for the next message pass.
// ---------------------------------------------------------------------------
__device__ __forceinline__ float sigmoidf_(float x) {
    return 1.0f / (1.0f + __expf(-x));
}

__global__ __launch_bounds__(256) void ggnn_gru(
    float* __restrict__ h, float* __restrict__ m,
    const float* __restrict__ W_ih, const float* __restrict__ W_hh,
    const float* __restrict__ b_ih, const float* __restrict__ b_hh, int Nn) {
    __shared__ float sWi[3 * HH * HH];
    __shared__ float sWh[3 * HH * HH];
    __shared__ float sbi[3 * HH];
    __shared__ float sbh[3 * HH];
    for (int i = threadIdx.x; i < 3 * HH * HH; i += blockDim.x) {
        sWi[i] = W_ih[i];
        sWh[i] = W_hh[i];
    }
    for (int i = threadIdx.x; i < 3 * HH; i += blockDim.x) {
        sbi[i] = b_ih[i];
        sbh[i] = b_hh[i];
    }
    __syncthreads();
    int v = blockIdx.x * blockDim.x + threadIdx.x;
    if (v >= Nn) return;

    float mv[HH], hv[HH], hn[HH];
    const float4* mp = (const float4*)(m + (size_t)v * HH);
    const float4* hp = (const float4*)(h + (size_t)v * HH);
    #pragma unroll
    for (int q = 0; q < 8; ++q) {
        float4 a = mp[q], b = hp[q];
        mv[4*q+0] = a.x; mv[4*q+1] = a.y; mv[4*q+2] = a.z; mv[4*q+3] = a.w;
        hv[4*q+0] = b.x; hv[4*q+1] = b.y; hv[4*q+2] = b.z; hv[4*q+3] = b.w;
    }
    #pragma unroll 4
    for (int j = 0; j < HH; ++j) {
        float gri = sbi[j],        grh = sbh[j];
        float gzi = sbi[HH + j],   gzh = sbh[HH + j];
        float gni = sbi[2*HH + j], gnh = sbh[2*HH + j];
        #pragma unroll
        for (int k = 0; k < HH; ++k) {
            gri = fmaf(sWi[j * HH + k],          mv[k], gri);
            grh = fmaf(sWh[j * HH + k],          hv[k], grh);
            gzi = fmaf(sWi[(HH + j) * HH + k],   mv[k], gzi);
            gzh = fmaf(sWh[(HH + j) * HH + k],   hv[k], gzh);
            gni = fmaf(sWi[(2*HH + j) * HH + k], mv[k], gni);
            gnh = fmaf(sWh[(2*HH + j) * HH + k], hv[k], gnh);
        }
        float rr = sigmoidf_(gri + grh);
        float zz = sigmoidf_(gzi + gzh);
        float nn = tanhf(gni + rr * gnh);
        hn[j] = (1.0f - zz) * nn + zz * hv[j];
    }
    float4* ho = (float4*)(h + (size_t)v * HH);
    float4* mo = (float4*)(m + (size_t)v * HH);
    #pragma unroll
    for (int q = 0; q < 8; ++q) {
        float4 t; t.x = hn[4*q+0]; t.y = hn[4*q+1]; t.z = hn[4*q+2]; t.w = hn[4*q+3];
        ho[q] = t;
        float4 z4; z4.x = z4.y = z4.z = z4.w = 0.0f;
        mo[q] = z4;   // reset accumulator for next step
    }
}

// ---------------------------------------------------------------------------
extern "C" void kernel_launch(void* const* d_in, const int* in_sizes, int n_in,
                              void* d_out, int out_size, void* d_ws, size_t ws_size,
                              hipStream_t stream) {
    const float* node_feat  = (const float*)d_in[0];
    const int*   edge_index = (const int*)  d_in[1];
    const float* edge_feat  = (const float*)d_in[2];
    const float* W_np       = (const float*)d_in[3];
    const float* b_np       = (const float*)d_in[4];
    const float* W_e        = (const float*)d_in[5];
    const float* b_e        = (const float*)d_in[6];
    const float* W_ih       = (const float*)d_in[7];
    const float* W_hh       = (const float*)d_in[8];
    const float* b_ih       = (const float*)d_in[9];
    const float* b_hh       = (const float*)d_in[10];
    float* h = (float*)d_out;

    const int Nn = in_sizes[0] / 64;     // 20000
    const int Ee = in_sizes[2] / EDIM;   // 160000
    const int* src = edge_index;
    const int* dst = edge_index + Ee;

    char* ws = (char*)d_ws;
    float2* bfrag = (float2*)ws;
    float*  mbuf  = (float*)(ws + BFRAG_BYTES);

    ggnn_build_bfrag<<<(BFRAG_ELEMS + 255) / 256, 256, 0, stream>>>(W_e, b_e, bfrag);
    ggnn_node_proj<<<(Nn + 255) / 256, 256, 0, stream>>>(node_feat, W_np, b_np, h, Nn);
    ggnn_zero_m<<<(Nn * HH + 255) / 256, 256, 0, stream>>>(mbuf, Nn * HH);

    const int tiles  = (Ee + 15) / 16;           // 10000
    const int wblk   = 256 / 32;                 // waves per block
    const int mblocks = (tiles + wblk - 1) / wblk;
    for (int step = 0; step < 3; ++step) {
        ggnn_message<<<mblocks, 256, 0, stream>>>(h, edge_feat, src, dst,
                                                  (const float2*)bfrag, mbuf, Ee);
        ggnn_gru<<<(Nn + 255) / 256, 256, 0, stream>>>(h, mbuf, W_ih, W_hh,
                                                       b_ih, b_hh, Nn);
    }
}